// ODEBlock_15178414424763
// MI455X (gfx1250) — compile-verified
//
#include <hip/hip_runtime.h>

typedef __attribute__((ext_vector_type(2))) float v2f;
typedef __attribute__((ext_vector_type(8))) float v8f;

#define N_STEPS 49
#define ALPHA_SL 0.001f

// 16 rows per wave, 8 waves per 256-thread block -> 128 rows per block.
// h tile kept in registers across all 49 Euler steps; W pre-staged in
// registers as WMMA B-fragments; per-step D->A relayout via wave-private LDS.
__global__ __launch_bounds__(256) void ode_euler_wmma_f32(
    const float* __restrict__ x, const float* __restrict__ W,
    const float* __restrict__ b, float* __restrict__ out, int nrows)
{
    // 8 waves * 16 rows * 68 floats (stride 68 breaks LDS bank conflicts)
    __shared__ float hsh[8 * 16 * 68];

    const int lane  = threadIdx.x & 31;
    const int wave  = threadIdx.x >> 5;
    const int lhalf = lane >> 4;   // 0 for lanes 0-15, 1 for lanes 16-31
    const int lmod  = lane & 15;
    const long row0 = (long)blockIdx.x * 128 + wave * 16;
    if (row0 + 16 > nrows) return;

    float* hws = &hsh[wave * 16 * 68];

    // ---- Stage W once into registers as B-layout fragments.
    // B-frag (kc, nb): lane l holds B[K = 2*lhalf + {0,1}][N = lmod]
    //                = W[16*nb + lmod][4*kc + 2*lhalf + {0,1}]
    v2f wfrag[16][4];
#pragma unroll
    for (int kc = 0; kc < 16; ++kc) {
#pragma unroll
        for (int nb = 0; nb < 4; ++nb) {
            const float* p = W + (16 * nb + lmod) * 64 + 4 * kc + 2 * lhalf;
            wfrag[kc][nb] = *(const v2f*)p;   // 8B-aligned
        }
    }

    // Bias: C/D element (M, N=lmod) in block nb needs b[16*nb + lmod]
    float breg[4];
#pragma unroll
    for (int nb = 0; nb < 4; ++nb) breg[nb] = b[16 * nb + lmod];

    // ---- Load h tile in C/D layout: h[nb][v] = x[row0 + v + 8*lhalf][16*nb + lmod]
    v8f h[4];
#pragma unroll
    for (int nb = 0; nb < 4; ++nb) {
#pragma unroll
        for (int v = 0; v < 8; ++v)
            h[nb][v] = x[(row0 + v + 8 * lhalf) * 64 + 16 * nb + lmod];
    }

    const float dt = 1.0f / 49.0f;
    const float c_lin = dt * ALPHA_SL;           // dt * alpha
    const float c_sp  = dt * (1.0f - ALPHA_SL);  // dt * (1 - alpha)

    // LDS store address (D-layout): row = v + 8*lhalf, col = 16*nb + lmod
    const int stbase = (8 * lhalf) * 68 + lmod;     // + v*68 + 16*nb
    // LDS load address (A-layout):  row = lmod, col = kc*4 + 2*lhalf (+{0,1})
    const int ldbase = lmod * 68 + 2 * lhalf;       // + kc*4

#pragma unroll 1
    for (int s = 0; s < N_STEPS; ++s) {
        // ---- spill h to wave-private LDS (D-layout rows/cols)
#pragma unroll
        for (int nb = 0; nb < 4; ++nb) {
#pragma unroll
            for (int v = 0; v < 8; ++v)
                hws[stbase + v * 68 + 16 * nb] = h[nb][v];
        }

        // ---- z = h @ W^T + b  via 64 chained fp32 WMMAs
        v8f z[4];
#pragma unroll
        for (int nb = 0; nb < 4; ++nb) {
            v8f c;
#pragma unroll
            for (int v = 0; v < 8; ++v) c[v] = breg[nb];
            z[nb] = c;
        }

#pragma unroll
        for (int kc = 0; kc < 16; ++kc) {
            // A-frag for K chunk kc (b64 load fills both A VGPRs per lane)
            v2f a = *(const v2f*)&hws[ldbase + kc * 4];
#pragma unroll
            for (int nb = 0; nb < 4; ++nb) {
                z[nb] = __builtin_amdgcn_wmma_f32_16x16x4_f32(
                    /*neg_a=*/false, a,
                    /*neg_b=*/false, wfrag[kc][nb],
                    /*c_mod=*/(short)0, z[nb],
                    /*reuse_a=*/false, /*reuse_b=*/false);
            }
        }

        // ---- h += dt*alpha*z + dt*(1-alpha)*softplus(z)
        // stable softplus: max(z,0) + log(1 + exp(-|z|)); the log1p->log(1+e)
        // substitution only matters for e < 2^-24 where the term is < 6e-8.
#pragma unroll
        for (int nb = 0; nb < 4; ++nb) {
#pragma unroll
            for (int v = 0; v < 8; ++v) {
                float zz = z[nb][v];
                float e  = __expf(-fabsf(zz));       // v_exp_f32
                float sp = fmaxf(zz, 0.0f) + __logf(1.0f + e);  // v_log_f32
                h[nb][v] = __builtin_fmaf(c_sp, sp,
                           __builtin_fmaf(c_lin, zz, h[nb][v]));
            }
        }
    }

    // ---- store result
#pragma unroll
    for (int nb = 0; nb < 4; ++nb) {
#pragma unroll
        for (int v = 0; v < 8; ++v)
            out[(row0 + v + 8 * lhalf) * 64 + 16 * nb + lmod] = h[nb][v];
    }
}

extern "C" void kernel_launch(void* const* d_in, const int* in_sizes, int n_in,
                              void* d_out, int out_size, void* d_ws, size_t ws_size,
                              hipStream_t stream) {
    const float* x = (const float*)d_in[0];   // [N, 64]
    const float* W = (const float*)d_in[1];   // [64, 64]
    const float* b = (const float*)d_in[2];   // [64]
    float* out = (float*)d_out;               // [N, 64]

    const int nrows = in_sizes[0] / 64;
    const int blocks = (nrows + 127) / 128;
    ode_euler_wmma_f32<<<blocks, 256, 0, stream>>>(x, W, b, out, nrows);
}